// SuperpixelEbli_53291954208855
// MI455X (gfx1250) — compile-verified
//
#include <hip/hip_runtime.h>
#include <hip/hip_bf16.h>

typedef float v2f __attribute__((ext_vector_type(2)));
typedef float v8f __attribute__((ext_vector_type(8)));

#define NEG_SLOPE 0.01f

// ---------------------------------------------------------------- zero fill
__global__ __launch_bounds__(256) void zero_kernel(float* __restrict__ p, long n) {
    long i = (long)blockIdx.x * blockDim.x + threadIdx.x;
    long stride = (long)gridDim.x * blockDim.x;
    for (; i < n; i += stride) p[i] = 0.0f;
}

// ---------------------------------------------------------------- SPMM (COO scatter)
// One wave32 per edge; lane = feature column (F == 32).
// Edge triples forced scalar (SMEM) via readfirstlane; vector path does one
// coalesced 128B gather of x[col][:] and one 128B batch of HW fp32 atomics.
__global__ __launch_bounds__(256) void spmm_kernel(
    const int* __restrict__ rows, const int* __restrict__ cols,
    const float* __restrict__ vals, const float* __restrict__ X,
    float* __restrict__ Y, int E)
{
    int e = __builtin_amdgcn_readfirstlane((blockIdx.x * 256 + threadIdx.x) >> 5);
    int lane = threadIdx.x & 31;
    if (e >= E) return;
    int   r = rows[e];
    int   c = cols[e];
    float v = vals[e];
    float xv = X[(long)c * 32 + lane];
    unsafeAtomicAdd(&Y[(long)r * 32 + lane], v * xv);
}

// ---------------------------------------------------------------- SCN GEMM via WMMA f32
// H[N,COUT] = act( concat(A[N,32], Y[N,32]) @ W[64,COUT] + b )
// One wave per 16-row tile; K=64 -> 16 x V_WMMA_F32_16X16X4_F32 per N-tile.
// A-frag (16x4 f32): lane(0-15)=M ; v0 = K=kb+2*hi, v1 = K=kb+2*hi+1
// B-frag (4x16 f32): lane%16 = N ; v0/v1 = W[kb+2*hi][n], W[kb+2*hi+1][n]
//   -> W staged TRANSPOSED in LDS as sWt[n][k] (zero-padded n) so each B-frag
//      is a single aligned ds_load_b64 at a compile-time immediate offset.
// D (16x16 f32): vgpr j -> M = j + 8*hi, N = lane%16.
template<int COUT, int LEAKY>
__global__ __launch_bounds__(256) void scn_gemm_kernel(
    const float* __restrict__ Ain, const float* __restrict__ Yin,
    const float* __restrict__ W,   const float* __restrict__ bias,
    float* __restrict__ Hout, int N)
{
    constexpr int COLS = (COUT + 15) & ~15;   // 32 or 16 (zero-padded)
    __shared__ float sWt[COLS * 64];          // [n][k], k-contiguous

    int tid = threadIdx.x;
    for (int idx = tid; idx < COLS * 64; idx += 256) {
        int n = idx >> 6;       // 0..COLS-1
        int k = idx & 63;       // 0..63
        sWt[idx] = (n < COUT) ? W[k * COUT + n] : 0.0f;
    }
    __syncthreads();

    int lane = tid & 31;
    int hi   = lane >> 4;
    int lo   = lane & 15;
    int tile = blockIdx.x * 8 + (tid >> 5);
    int row0 = tile * 16;
    if (row0 >= N) return;  // wave-uniform: surviving waves keep full EXEC for WMMA

    v8f acc0 = {0.f,0.f,0.f,0.f,0.f,0.f,0.f,0.f};
    v8f acc1 = {0.f,0.f,0.f,0.f,0.f,0.f,0.f,0.f};

    // Clamp OOB rows: garbage in A-row m only affects D-row m, which is never stored.
    int  r  = row0 + lo;
    long rb = (long)((r < N) ? r : (N - 1)) * 32 + 2 * hi;

    const float* __restrict__ pA = Ain + rb;
    const float* __restrict__ pY = Yin + rb;
    const float* __restrict__ pB0 = sWt + (lo * 64 + 2 * hi);            // ntile 0
    const float* __restrict__ pB1 = sWt + ((16 + lo) * 64 + 2 * hi);     // ntile 1

    #pragma unroll
    for (int kt = 0; kt < 16; ++kt) {
        int kb = kt * 4;
        const float* __restrict__ src = (kb < 32) ? pA : pY;
        v2f a = *(const v2f*)(src + (kb & 31));          // global_load_b64, imm offset
        v2f b0 = *(const v2f*)(pB0 + kb);                // ds_load_b64, imm offset
        acc0 = __builtin_amdgcn_wmma_f32_16x16x4_f32(
            false, a, false, b0, (short)0, acc0, false, false);
        if (COUT > 16) {
            v2f b1 = *(const v2f*)(pB1 + kb);
            acc1 = __builtin_amdgcn_wmma_f32_16x16x4_f32(
                false, a, false, b1, (short)0, acc1, false, false);
        }
    }

    float bs0 = (lo < COUT) ? bias[lo] : 0.0f;
    float bs1 = (COUT > 16) ? bias[16 + lo] : 0.0f;

    #pragma unroll
    for (int j = 0; j < 8; ++j) {
        int m  = j + 8 * hi;
        int rr = row0 + m;
        if (rr >= N) continue;
        if (lo < COUT) {
            float v = acc0[j] + bs0;
            if (LEAKY) v = (v > 0.f) ? v : NEG_SLOPE * v;
            Hout[(long)rr * COUT + lo] = v;
        }
        if (COUT > 16) {
            float v = acc1[j] + bs1;
            if (LEAKY) v = (v > 0.f) ? v : NEG_SLOPE * v;
            Hout[(long)rr * COUT + 16 + lo] = v;
        }
    }
}

// ---------------------------------------------------------------- segment mean-pool (sum phase)
__global__ __launch_bounds__(256) void pool_sum_kernel(
    const float* __restrict__ H, const int* __restrict__ batch,
    float* __restrict__ pooled /* [G,30] */, float* __restrict__ cnt /* [G] */,
    int N, int colOff)
{
    int i = blockIdx.x * blockDim.x + threadIdx.x;
    if (i >= N) return;
    int g = batch[i];
    #pragma unroll
    for (int j = 0; j < 10; ++j)
        unsafeAtomicAdd(&pooled[(long)g * 30 + colOff + j], H[(long)i * 10 + j]);
    unsafeAtomicAdd(&cnt[g], 1.0f);
}

// ---------------------------------------------------------------- final head: mean, [G,30]@[30,10]+b, softmax
__global__ __launch_bounds__(256) void final_kernel(
    const float* __restrict__ pooled, const float* __restrict__ cnt,
    const float* __restrict__ Wf, const float* __restrict__ bf,
    float* __restrict__ out, int G)
{
    int g = blockIdx.x * blockDim.x + threadIdx.x;
    if (g >= G) return;

    float feat[30];
    #pragma unroll
    for (int b = 0; b < 3; ++b) {
        float c = cnt[b * G + g];
        c = (c > 1.f) ? c : 1.f;
        float inv = 1.f / c;
        #pragma unroll
        for (int j = 0; j < 10; ++j)
            feat[b * 10 + j] = pooled[(long)g * 30 + b * 10 + j] * inv;
    }

    float logit[10];
    float mx = -1e30f;
    #pragma unroll
    for (int o = 0; o < 10; ++o) {
        float s = bf[o];
        for (int k = 0; k < 30; ++k) s += feat[k] * Wf[k * 10 + o];
        logit[o] = s;
        mx = fmaxf(mx, s);
    }
    float sum = 0.f;
    #pragma unroll
    for (int o = 0; o < 10; ++o) { logit[o] = expf(logit[o] - mx); sum += logit[o]; }
    float inv = 1.f / sum;
    #pragma unroll
    for (int o = 0; o < 10; ++o) out[(long)g * 10 + o] = logit[o] * inv;
}

// ---------------------------------------------------------------- host
static inline void launch_zero(float* p, long n, hipStream_t s) {
    long blocks = (n + 255) / 256;
    if (blocks > 4096) blocks = 4096;
    if (blocks < 1) blocks = 1;
    zero_kernel<<<(int)blocks, 256, 0, s>>>(p, n);
}

extern "C" void kernel_launch(void* const* d_in, const int* in_sizes, int n_in,
                              void* d_out, int out_size, void* d_ws, size_t ws_size,
                              hipStream_t stream) {
    // setup_inputs() dict order
    const float* x[3]    = {(const float*)d_in[0], (const float*)d_in[1], (const float*)d_in[2]};
    const float* vals[3] = {(const float*)d_in[3], (const float*)d_in[4], (const float*)d_in[5]};
    const int*   rows[3] = {(const int*)d_in[6],  (const int*)d_in[8],  (const int*)d_in[10]};
    const int*   cols[3] = {(const int*)d_in[7],  (const int*)d_in[9],  (const int*)d_in[11]};
    const int*   batch[3]= {(const int*)d_in[12], (const int*)d_in[13], (const int*)d_in[14]};
    // d_in[15] = num_graphs (device scalar) — G derived from out_size instead (host-visible).

    int N[3] = {in_sizes[0] / 32, in_sizes[1] / 32, in_sizes[2] / 32};
    int E[3] = {in_sizes[3], in_sizes[4], in_sizes[5]};

    const float *W1[3], *b1[3], *W2[3], *b2[3], *W3[3], *b3[3];
    for (int d = 0; d < 3; ++d) {
        int base = 16 + d * 6;
        W1[d] = (const float*)d_in[base + 0]; b1[d] = (const float*)d_in[base + 1];
        W2[d] = (const float*)d_in[base + 2]; b2[d] = (const float*)d_in[base + 3];
        W3[d] = (const float*)d_in[base + 4]; b3[d] = (const float*)d_in[base + 5];
    }
    const float* Wf = (const float*)d_in[34];
    const float* bf = (const float*)d_in[35];

    int G = out_size / 10;
    int Nmax = N[0];
    if (N[1] > Nmax) Nmax = N[1];
    if (N[2] > Nmax) Nmax = N[2];

    // workspace layout (floats)
    float* ws     = (float*)d_ws;
    size_t nm32   = (size_t)Nmax * 32;
    float* Y      = ws;                       // spmm accumulator  [Nmax,32]
    float* A      = Y + nm32;                 // activation ping   [Nmax,32]
    float* H      = A + nm32;                 // activation pong   [Nmax,32]
    float* pooled = H + nm32;                 // [G,30]
    float* cnt    = pooled + (size_t)G * 30;  // [3,G]

    // zero pooled + counts (contiguous)
    launch_zero(pooled, (long)G * 30 + 3L * G, stream);

    for (int d = 0; d < 3; ++d) {
        int n = N[d], e = E[d];
        int spmmBlocks = (e + 7) / 8;               // 8 waves/block, 1 edge/wave
        int gemmBlocks = ((n + 15) / 16 + 7) / 8;   // 8 tiles/block, 16 rows/tile
        int poolBlocks = (n + 255) / 256;

        // layer 1:  a1 = leaky([x, L@x] @ W1 + b1)   -> A
        launch_zero(Y, (long)n * 32, stream);
        spmm_kernel<<<spmmBlocks, 256, 0, stream>>>(rows[d], cols[d], vals[d], x[d], Y, e);
        scn_gemm_kernel<32,1><<<gemmBlocks, 256, 0, stream>>>(x[d], Y, W1[d], b1[d], A, n);

        // layer 2:  a2 = leaky([a1, L@a1] @ W2 + b2) -> H
        launch_zero(Y, (long)n * 32, stream);
        spmm_kernel<<<spmmBlocks, 256, 0, stream>>>(rows[d], cols[d], vals[d], A, Y, e);
        scn_gemm_kernel<32,1><<<gemmBlocks, 256, 0, stream>>>(A, Y, W2[d], b2[d], H, n);

        // layer 3:  h3 = [a2, L@a2] @ W3 + b3        -> A (N x 10)
        launch_zero(Y, (long)n * 32, stream);
        spmm_kernel<<<spmmBlocks, 256, 0, stream>>>(rows[d], cols[d], vals[d], H, Y, e);
        scn_gemm_kernel<10,0><<<gemmBlocks, 256, 0, stream>>>(H, Y, W3[d], b3[d], A, n);

        // mean-pool sums into pooled[:, d*10 : d*10+10]
        pool_sum_kernel<<<poolBlocks, 256, 0, stream>>>(A, batch[d], pooled, cnt + (size_t)d * G, n, d * 10);
    }

    final_kernel<<<(G + 255) / 256, 256, 0, stream>>>(pooled, cnt, Wf, bf, (float*)d_out, G);
}